// ColorLoss_86225763435418
// MI455X (gfx1250) — compile-verified
//
#include <hip/hip_runtime.h>
#include <math.h>

typedef float v2f __attribute__((ext_vector_type(2)));
typedef float v8f __attribute__((ext_vector_type(8)));

#define THREADS_PER_WG 256
#define WAVES_PER_WG   8          // wave32 on gfx1250
#define ROWS_PER_WAVE  32         // two 16-row A fragments per wave
#define ROWS_PER_WG    (WAVES_PER_WG * ROWS_PER_WAVE)   // 256 pred rows per WG
#define CHUNK_PTS      1024       // GT points staged per chunk (16 KB LDS)
#define TILES_PER_CHUNK (CHUNK_PTS / 16)

// Each wave owns 32 pred rows as two A fragments {p0,p1,p2,1}; B columns are
// {g0,g1,g2,-||g||^2/2} so each V_WMMA_F32_16X16X4_F32 yields
//   dot(p,g) - ||g||^2/2
// directly (the epilogue add is folded into the K=3 pad channel). Then
//   min_j ||p-g||^2 = ||p||^2 - 2 * max_j(dot - ||g||^2/2).
// LDS holds B fragments pre-swizzled into the ISA 4x16 lane layout; every
// staged fragment feeds two WMMAs (one per A fragment).
__global__ __launch_bounds__(THREADS_PER_WG)
void chamfer_color_wmma(const float* __restrict__ pred,
                        const float* __restrict__ gt,
                        float* __restrict__ wg_partials,
                        int M, int N) {
  // per tile: 32 lanes x float2  (lane c: {g0,g1}=K0/K1, lane c+16: {g2,-g2/2}=K2/K3)
  __shared__ float2 sB[TILES_PER_CHUNK * 32];
  __shared__ float  sWave[WAVES_PER_WG];

  const int tid  = threadIdx.x;
  const int lane = tid & 31;
  const int wave = tid >> 5;
  const int half = lane >> 4;
  const int l16  = lane & 15;

  const int wgPerBatch = (M + ROWS_PER_WG - 1) / ROWS_PER_WG;
  const int batch   = blockIdx.x / wgPerBatch;
  const int rowBase = (blockIdx.x % wgPerBatch) * ROWS_PER_WG + wave * ROWS_PER_WAVE;

  const float* predB = pred + (size_t)batch * M * 3;
  const float* gtB   = gt   + (size_t)batch * N * 3;

  // ---- Two A fragments: rows [rowBase, rowBase+16) and [rowBase+16, rowBase+32).
  // ISA 16x4 layout: VGPR0 = K0 (lanes 0-15) / K2 (lanes 16-31); VGPR1 = K1 / K3.
  v2f afragA, afragB;
  {
    int r0 = rowBase + l16;
    int r1 = rowBase + 16 + l16;
    int c0 = (r0 < M) ? r0 : (M - 1);
    int c1 = (r1 < M) ? r1 : (M - 1);
    float a0 = predB[c0 * 3 + 0], a1 = predB[c0 * 3 + 1], a2 = predB[c0 * 3 + 2];
    float b0 = predB[c1 * 3 + 0], b1 = predB[c1 * 3 + 1], b2 = predB[c1 * 3 + 2];
    afragA.x = half ? a2   : a0;
    afragA.y = half ? 1.0f : a1;   // K=3 of A is the constant 1
    afragB.x = half ? b2   : b0;
    afragB.y = half ? 1.0f : b1;
  }

  // Running max of (dot - ||g||^2/2); D layout: VGPR i = rows i / i+8.
  v8f mx0, mx1;
#pragma unroll
  for (int i = 0; i < 8; ++i) { mx0[i] = -3.0e38f; mx1[i] = -3.0e38f; }

  const int nChunks = (N + CHUNK_PTS - 1) / CHUNK_PTS;
  for (int ch = 0; ch < nChunks; ++ch) {
    const int base = ch * CHUNK_PTS;
    // ---- stage: 256 threads x 4 points, written pre-swizzled as B fragments ----
#pragma unroll
    for (int k = 0; k < CHUNK_PTS / THREADS_PER_WG; ++k) {
      int idx  = k * THREADS_PER_WG + tid;
      int gidx = base + idx;
      float x = 0.0f, y = 0.0f, z = 0.0f, w = -3.0e38f;  // pad: can never win max
      if (gidx < N) {
        x = gtB[gidx * 3 + 0];
        y = gtB[gidx * 3 + 1];
        z = gtB[gidx * 3 + 2];
        w = -0.5f * (x * x + y * y + z * z);
      }
      int t = idx >> 4;
      int c = idx & 15;
      sB[t * 32 + c]      = make_float2(x, y);   // lanes 0-15: K0,K1
      sB[t * 32 + 16 + c] = make_float2(z, w);   // lanes 16-31: K2,K3
    }
    __syncthreads();

    // speculative L2 prefetch of the next chunk (global_prefetch_b8)
    if (ch + 1 < nChunks) {
      int pidx = (ch + 1) * CHUNK_PTS + tid * (CHUNK_PTS / THREADS_PER_WG);
      if (pidx < N) __builtin_prefetch(gtB + (size_t)pidx * 3, 0, 1);
    }

    // ---- compute: per tile = 1 shared ds_load_b64 + 2 wmma + 8 max3 ----
#pragma unroll 4
    for (int t = 0; t < TILES_PER_CHUNK; ++t) {
      float2 b = sB[t * 32 + lane];              // fragment ready-made in LDS
      v2f bfrag;
      bfrag.x = b.x;
      bfrag.y = b.y;
      v8f czero = {};
      v8f d0 = __builtin_amdgcn_wmma_f32_16x16x4_f32(
          false, afragA, false, bfrag, (short)0, czero, false, false);
      v8f d1 = __builtin_amdgcn_wmma_f32_16x16x4_f32(
          false, afragB, false, bfrag, (short)0, czero, false, false);
#pragma unroll
      for (int i = 0; i < 8; ++i) {
        mx0[i] = fmaxf(mx0[i], d0[i]);
        mx1[i] = fmaxf(mx1[i], d1[i]);
      }
    }
    __syncthreads();
  }

  // ---- max across the 16 columns of each half (xor stays inside each half) ----
#pragma unroll
  for (int m = 8; m >= 1; m >>= 1) {
#pragma unroll
    for (int i = 0; i < 8; ++i) {
      mx0[i] = fmaxf(mx0[i], __shfl_xor(mx0[i], m, 32));
      mx1[i] = fmaxf(mx1[i], __shfl_xor(mx1[i], m, 32));
    }
  }

  // lanes 0 and 16 finalize 8 rows of each fragment: dist = sqrt(||p||^2 - 2*max)
  float wsum = 0.0f;
  if (l16 == 0) {
#pragma unroll
    for (int i = 0; i < 8; ++i) {
      int row0 = rowBase + half * 8 + i;         // rows of fragment A
      if (row0 < M) {
        float x = predB[row0 * 3 + 0];
        float y = predB[row0 * 3 + 1];
        float z = predB[row0 * 3 + 2];
        float d2 = x * x + y * y + z * z - 2.0f * mx0[i];
        d2 = d2 > 0.0f ? d2 : 0.0f;
        wsum += sqrtf(d2);
      }
      int row1 = rowBase + 16 + half * 8 + i;    // rows of fragment B
      if (row1 < M) {
        float x = predB[row1 * 3 + 0];
        float y = predB[row1 * 3 + 1];
        float z = predB[row1 * 3 + 2];
        float d2 = x * x + y * y + z * z - 2.0f * mx1[i];
        d2 = d2 > 0.0f ? d2 : 0.0f;
        wsum += sqrtf(d2);
      }
    }
  }
  wsum += __shfl_xor(wsum, 16, 32);              // lane0 holds full wave sum
  if (lane == 0) sWave[wave] = wsum;
  __syncthreads();
  if (tid == 0) {
    float s = 0.0f;
#pragma unroll
    for (int w = 0; w < WAVES_PER_WG; ++w) s += sWave[w];
    wg_partials[blockIdx.x] = s;
  }
}

// Deterministic final reduction: fixed-order strided sums + LDS tree.
__global__ __launch_bounds__(256)
void chamfer_reduce(const float* __restrict__ partials, float* __restrict__ out,
                    int nPartials, float scale) {
  __shared__ float s[256];
  int t = threadIdx.x;
  float a = 0.0f;
  for (int i = t; i < nPartials; i += 256) a += partials[i];
  s[t] = a;
  __syncthreads();
  for (int step = 128; step > 0; step >>= 1) {
    if (t < step) s[t] += s[t + step];
    __syncthreads();
  }
  if (t == 0) out[0] = s[0] * scale;
}

extern "C" void kernel_launch(void* const* d_in, const int* in_sizes, int n_in,
                              void* d_out, int out_size, void* d_ws, size_t ws_size,
                              hipStream_t stream) {
  const float* pred = (const float*)d_in[0];   // [2, M, 3] f32
  const float* gt   = (const float*)d_in[1];   // [2, N, 3] f32
  const int B = 2;
  const int M = in_sizes[0] / (B * 3);
  const int N = in_sizes[1] / (B * 3);

  float* partials = (float*)d_ws;
  const int wgPerBatch = (M + ROWS_PER_WG - 1) / ROWS_PER_WG;
  const int nBlocks = B * wgPerBatch;

  chamfer_color_wmma<<<nBlocks, THREADS_PER_WG, 0, stream>>>(pred, gt, partials, M, N);

  const float scale = 1.0f / (float)((size_t)B * (size_t)M);  // mean * LOSS_WEIGHT(=1)
  chamfer_reduce<<<1, 256, 0, stream>>>(partials, (float*)d_out, nBlocks, scale);
}